// HandshakingKernel_15290083574191
// MI455X (gfx1250) — compile-verified
//
#include <hip/hip_runtime.h>
#include <cstdint>

#define B_   8
#define S_   256
#define D_   768
#define VF_  32
#define P_   7696          // sum over i of min(S-i, VF)
#define MBLK 48            // pairs per block in k_main
#define NT3  161           // ceil(P_/MBLK)
#define NW3  12            // waves per block in k_main (384 threads)
#define EPSF 1e-12f
#define XSTR 772           // 768 + 4 pad (even -> float2-aligned, distinct banks)

typedef float v2f __attribute__((ext_vector_type(2)));
typedef float v8f __attribute__((ext_vector_type(8)));

__device__ __forceinline__ float wave_sum(float v) {
#pragma unroll
  for (int off = 16; off > 0; off >>= 1) v += __shfl_xor(v, off, 32);
  return v;
}

// ---------------- K0: pair index tables -------------------------------------
__global__ void k_pairs(int* __restrict__ pairI, int* __restrict__ pairJ) {
  int i = threadIdx.x;           // 256 threads, one per i
  if (i >= S_) return;
  int cnt = (S_ - i < VF_) ? (S_ - i) : VF_;
  int off;
  if (i <= S_ - VF_) {
    off = VF_ * i;
  } else {
    int c = S_ - i;                                // tail count for this i
    off = (S_ - VF_ + 1) * VF_ + (496 - c * (c + 1) / 2);
  }
  for (int j = 0; j < cnt; ++j) { pairI[off + j] = i; pairJ[off + j] = j; }
}

// ---------------- K1: cumsum over sequence dim ------------------------------
__global__ void k_cumsum(const float* __restrict__ h, float* __restrict__ cs) {
  int idx = blockIdx.x * blockDim.x + threadIdx.x;   // B_*D_ columns
  if (idx >= B_ * D_) return;
  int b = idx / D_, d = idx - b * D_;
  const float* hp = h  + (size_t)b * S_ * D_ + d;
  float*       cp = cs + (size_t)b * (S_ + 1) * D_ + d;
  float acc = 0.f;
  cp[0] = 0.f;
  for (int s = 0; s < S_; ++s) {
    acc += hp[(size_t)s * D_];
    cp[(size_t)(s + 1) * D_] = acc;
  }
}

// ---------------- K2: LN1 projections (2048 rows) via f32 WMMA --------------
__global__ void __launch_bounds__(256)
k_proj1(const float* __restrict__ h, const float* __restrict__ Wb,
        const float* __restrict__ Wg, const float* __restrict__ beta0,
        const float* __restrict__ gamma0,
        float* __restrict__ outB, float* __restrict__ outG) {
  __shared__ float Xs[32][XSTR];
  const int r0 = blockIdx.x * 32;
  for (int t = threadIdx.x; t < 32 * D_; t += 256) {
    int m = t / D_, d = t - m * D_;
    Xs[m][d] = h[(size_t)(r0 + m) * D_ + d];
  }
  __syncthreads();

  const int lane = threadIdx.x & 31, wv = threadIdx.x >> 5;
  const int l15 = lane & 15, hh = lane >> 4;
  for (int u = wv; u < 48; u += 8) {            // 48 N-tiles of 16
    const int n = u * 16 + l15;
    const float* wbp = Wb + (size_t)n * D_ + 2 * hh;
    const float* wgp = Wg + (size_t)n * D_ + 2 * hh;
    v8f ab0 = {}, ab1 = {}, ag0 = {}, ag1 = {};
    for (int k0 = 0; k0 < D_; k0 += 4) {
      v2f bb = *(const v2f*)(wbp + k0);
      v2f bg = *(const v2f*)(wgp + k0);
      v2f a0 = *(const v2f*)&Xs[l15][k0 + 2 * hh];
      v2f a1 = *(const v2f*)&Xs[16 + l15][k0 + 2 * hh];
      ab0 = __builtin_amdgcn_wmma_f32_16x16x4_f32(false, a0, false, bb, (short)0, ab0, false, false);
      ab1 = __builtin_amdgcn_wmma_f32_16x16x4_f32(false, a1, false, bb, (short)0, ab1, false, false);
      ag0 = __builtin_amdgcn_wmma_f32_16x16x4_f32(false, a0, false, bg, (short)0, ag0, false, false);
      ag1 = __builtin_amdgcn_wmma_f32_16x16x4_f32(false, a1, false, bg, (short)0, ag1, false, false);
    }
    const float b0v = beta0[n], g0v = gamma0[n];
#pragma unroll
    for (int r = 0; r < 8; ++r) {
      int m0 = r + 8 * hh;
      size_t row0 = (size_t)(r0 + m0) * D_ + n;
      size_t row1 = (size_t)(r0 + 16 + m0) * D_ + n;
      outB[row0] = ab0[r] + b0v;  outG[row0] = ag0[r] + g0v;
      outB[row1] = ab1[r] + b0v;  outG[row1] = ag1[r] + g0v;
    }
  }
}

// ---------------- K3: fused full_mean GEMM + double layer-norm --------------
__global__ void __launch_bounds__(NW3 * 32)
k_main(const float* __restrict__ h,  const float* __restrict__ cs,
       const float* __restrict__ p1b, const float* __restrict__ p1g,
       const float* __restrict__ Wb2, const float* __restrict__ Wg2,
       const float* __restrict__ beta2, const float* __restrict__ gamma2,
       const int* __restrict__ pairI, const int* __restrict__ pairJ,
       float* __restrict__ out) {
  __shared__ float Xs[MBLK][XSTR];                 // ~148 KB
  __shared__ int   pi[MBLK], pj[MBLK];
  __shared__ float muA[MBLK], ivA[MBLK], mu1[MBLK], iv1s[MBLK];

  const int tile = blockIdx.x % NT3;
  const int b    = blockIdx.x / NT3;
  const int p0   = tile * MBLK;

  if (threadIdx.x < MBLK) {
    int p = p0 + threadIdx.x;
    pi[threadIdx.x] = (p < P_) ? pairI[p] : 0;
    pj[threadIdx.x] = (p < P_) ? pairJ[p] : 0;
  }
  __syncthreads();

  const float* csb = cs + (size_t)b * (S_ + 1) * D_;

  // Phase 1: stage full_mean tile (MBLK x 768) into LDS
  for (int t = threadIdx.x; t < MBLK * D_; t += NW3 * 32) {
    int m = t / D_, d = t - m * D_;
    float v = 0.f;
    if (p0 + m < P_) {
      int i = pi[m], j = pj[m];
      v = (csb[(size_t)(i + j + 1) * D_ + d] - csb[(size_t)i * D_ + d]) * (1.0f / (float)(j + 1));
    }
    Xs[m][d] = v;
  }

  // Phase 2: per-row LN stats (wave32 reductions); 4 rows per wave
  const int lane = threadIdx.x & 31, wv = threadIdx.x >> 5;
  for (int m = wv * 4; m < wv * 4 + 4; ++m) {
    if (p0 + m >= P_) {
      if (lane == 0) { muA[m] = 0.f; ivA[m] = 0.f; mu1[m] = 0.f; iv1s[m] = 0.f; }
      continue;
    }
    int i = pi[m], j = pj[m];
    const float* aft = h   + ((size_t)b * S_ + (i + j)) * D_;
    const float* g1r = p1g + ((size_t)b * S_ + i) * D_;
    const float* b1r = p1b + ((size_t)b * S_ + i) * D_;
    float s = 0.f, s2 = 0.f;
    for (int d = lane; d < D_; d += 32) { float v = aft[d]; s += v; s2 += v * v; }
    s = wave_sum(s); s2 = wave_sum(s2);
    float mu  = s * (1.0f / (float)D_);
    float var = s2 * (1.0f / (float)D_) - mu * mu;
    float iv  = var + EPSF; iv = 1.0f / (iv * iv);       // reference: /(var+eps)^2
    float t1 = 0.f, t2 = 0.f;
    for (int d = lane; d < D_; d += 32) {
      float y = (aft[d] - mu) * iv * g1r[d] + b1r[d];
      t1 += y; t2 += y * y;
    }
    t1 = wave_sum(t1); t2 = wave_sum(t2);
    float m1 = t1 * (1.0f / (float)D_);
    float v1 = t2 * (1.0f / (float)D_) - m1 * m1;
    float j1 = v1 + EPSF; j1 = 1.0f / (j1 * j1);
    if (lane == 0) { muA[m] = mu; ivA[m] = iv; mu1[m] = m1; iv1s[m] = j1; }
  }
  __syncthreads();

  // Phase 3: WMMA GEMM (beta2/gamma2) + fused epilogue
  const int l15 = lane & 15, hh = lane >> 4;
  for (int u = wv; u < 48; u += NW3) {            // 4 N-tiles per wave
    const int n = u * 16 + l15;
    const float* wbp = Wb2 + (size_t)n * D_ + 2 * hh;
    const float* wgp = Wg2 + (size_t)n * D_ + 2 * hh;
    v8f ab[3], ag[3];
#pragma unroll
    for (int mt = 0; mt < 3; ++mt) { ab[mt] = {}; ag[mt] = {}; }
    for (int k0 = 0; k0 < D_; k0 += 4) {
      if ((k0 & 63) == 0) {                       // pull W 256B ahead (speculative)
        __builtin_prefetch(wbp + k0 + 64, 0, 3);
        __builtin_prefetch(wgp + k0 + 64, 0, 3);
      }
      v2f bb = *(const v2f*)(wbp + k0);
      v2f bg = *(const v2f*)(wgp + k0);
#pragma unroll
      for (int mt = 0; mt < 3; ++mt) {
        v2f a = *(const v2f*)&Xs[mt * 16 + l15][k0 + 2 * hh];
        ab[mt] = __builtin_amdgcn_wmma_f32_16x16x4_f32(false, a, false, bb, (short)0, ab[mt], false, false);
        ag[mt] = __builtin_amdgcn_wmma_f32_16x16x4_f32(false, a, false, bg, (short)0, ag[mt], false, false);
      }
    }
    const float b2v = beta2[n], g2v = gamma2[n];
#pragma unroll
    for (int mt = 0; mt < 3; ++mt) {
#pragma unroll
      for (int r = 0; r < 8; ++r) {
        int m = mt * 16 + r + 8 * hh;
        int p = p0 + m;
        if (p < P_) {
          int i = pi[m], j = pj[m];
          float aft = h[((size_t)b * S_ + (i + j)) * D_ + n];
          float g1  = p1g[((size_t)b * S_ + i) * D_ + n];
          float b1  = p1b[((size_t)b * S_ + i) * D_ + n];
          float y1  = (aft - muA[m]) * ivA[m] * g1 + b1;
          float yn  = (y1 - mu1[m]) * iv1s[m];
          float bet = ab[mt][r] + b2v;
          float gam = ag[mt][r] + g2v;
          out[((size_t)b * P_ + p) * D_ + n] = yn * gam + bet;
        }
      }
    }
  }
}

// ---------------- host launcher ---------------------------------------------
extern "C" void kernel_launch(void* const* d_in, const int* in_sizes, int n_in,
                              void* d_out, int out_size, void* d_ws, size_t ws_size,
                              hipStream_t stream) {
  (void)in_sizes; (void)n_in; (void)out_size; (void)ws_size;
  const float* h      = (const float*)d_in[0];
  const float* Wb1    = (const float*)d_in[1];
  const float* Wg1    = (const float*)d_in[2];
  const float* Wb2    = (const float*)d_in[3];
  const float* Wg2    = (const float*)d_in[4];
  const float* beta1  = (const float*)d_in[5];
  const float* gamma1 = (const float*)d_in[6];
  const float* beta2  = (const float*)d_in[7];
  const float* gamma2 = (const float*)d_in[8];
  float* out = (float*)d_out;

  size_t csN = (size_t)B_ * (S_ + 1) * D_;
  size_t pN  = (size_t)B_ * S_ * D_;
  float* cs   = (float*)d_ws;
  float* p1b  = cs + csN;
  float* p1g  = p1b + pN;
  int*   pI   = (int*)(p1g + pN);
  int*   pJ   = pI + P_;

  k_pairs <<<1, 256, 0, stream>>>(pI, pJ);
  k_cumsum<<<(B_ * D_ + 255) / 256, 256, 0, stream>>>(h, cs);
  k_proj1 <<<(B_ * S_) / 32, 256, 0, stream>>>(h, Wb1, Wg1, beta1, gamma1, p1b, p1g);
  k_main  <<<NT3 * B_, NW3 * 32, 0, stream>>>(h, cs, p1b, p1g, Wb2, Wg2, beta2, gamma2, pI, pJ, out);
}